// RankingModel_v2_25237227831810
// MI455X (gfx1250) — compile-verified
//
#include <hip/hip_runtime.h>
#include <hip/hip_bf16.h>
#include <math.h>

typedef float v2f __attribute__((ext_vector_type(2)));
typedef float v8f __attribute__((ext_vector_type(8)));

#define B_     32
#define ROWS   16384
#define FEAT   256
#define HID    16
#define LN_EPS 1e-5f
#define N_     ROWS
#define NT     1024

// ---------------------------------------------------------------------------
// Kernel 1: fused LN1 -> GEMM(256x16) via v_wmma_f32_16x16x4_f32 -> +b1, ReLU
//           -> LN2(16) -> GEMM(16x1) -> scores[B][ROWS]
// One wave (32 lanes) owns a 16-row tile; 4 waves / block -> 64 rows / block.
// Tile staged with CDNA5 async global->LDS copies (ASYNCcnt), no VGPR staging.
// ---------------------------------------------------------------------------
__global__ __launch_bounds__(128) void score_kernel(
    const float* __restrict__ table, const float* __restrict__ ln1_g,
    const float* __restrict__ ln1_b, const float* __restrict__ W1,
    const float* __restrict__ b1,    const float* __restrict__ ln2_g,
    const float* __restrict__ ln2_b, const float* __restrict__ W2,
    const float* __restrict__ b2,    float* __restrict__ scores)
{
    __shared__ float s_w1[FEAT * HID];     // 16 KB
    __shared__ float s_g[FEAT];
    __shared__ float s_bb[FEAT];
    __shared__ float s_tile[4][16 * FEAT]; // 64 KB (320KB/WGP on CDNA5)

    const int tid  = threadIdx.x;
    const int wave = tid >> 5;
    const int lane = tid & 31;

    const int tilesPerBatch = ROWS / 64;
    const int batch = blockIdx.x / tilesPerBatch;
    const int row0  = (blockIdx.x % tilesPerBatch) * 64;

    // --- async global->LDS tile copy (CDNA5 GLOBAL_LOAD_ASYNC_TO_LDS_B128) ---
    // 16 rows x 256 f32 = 16 KB per wave; 32 lanes x 16B = 512B per issue.
    {
        const int rowbase = row0 + wave * 16;
        const char* gsrc = (const char*)(table + ((size_t)batch * ROWS + rowbase) * FEAT);
        // generic shared pointer low 32 bits == LDS byte offset (flat aperture)
        unsigned ldsbase = (unsigned)(uintptr_t)(&s_tile[wave][0]);
        #pragma unroll 4
        for (int it = 0; it < 32; ++it) {
            unsigned ldsoff = ldsbase + (unsigned)(it * 512 + lane * 16);
            const char* ga  = gsrc + it * 512 + lane * 16;
            asm volatile("global_load_async_to_lds_b128 %0, %1, off"
                         :: "v"(ldsoff), "v"(ga) : "memory");
        }
    }

    for (int i = tid; i < FEAT * HID; i += 128) s_w1[i] = W1[i];
    for (int i = tid; i < FEAT; i += 128) { s_g[i] = ln1_g[i]; s_bb[i] = ln1_b[i]; }

    asm volatile("s_wait_asynccnt 0" ::: "memory");
    __syncthreads();

    float* t = &s_tile[wave][0];
    const int rowbase = row0 + wave * 16;

    // Lane mapping per ISA 16x4 f32 A-matrix layout:
    // lanes 0-15: rows 0-15 / K pair {0,1}; lanes 16-31: rows 0-15 / K pair {2,3}
    const int half = lane >> 4;
    const int row  = lane & 15;
    const int col  = lane & 15;   // N index for B/C/D fragments
    const float* tr = t + row * FEAT;

    // LN1 stats: each lane sums half of its row, combine halves with xor-16.
    float sum = 0.f, sq = 0.f;
    #pragma unroll 4
    for (int i = 0; i < 128; ++i) {
        float v = tr[half * 128 + i];
        sum += v; sq += v * v;
    }
    sum += __shfl_xor(sum, 16, 32);
    sq  += __shfl_xor(sq, 16, 32);
    const float mean = sum * (1.0f / FEAT);
    const float var  = sq * (1.0f / FEAT) - mean * mean;
    const float rstd = rsqrtf(var + LN_EPS);

    v8f acc = {0.f, 0.f, 0.f, 0.f, 0.f, 0.f, 0.f, 0.f};
    for (int k0 = 0; k0 < FEAT; k0 += 4) {
        const int kk = k0 + half * 2;
        v2f a, bf;
        a.x = (tr[kk]     - mean) * rstd * s_g[kk]     + s_bb[kk];
        a.y = (tr[kk + 1] - mean) * rstd * s_g[kk + 1] + s_bb[kk + 1];
        bf.x = s_w1[kk * HID + col];
        bf.y = s_w1[(kk + 1) * HID + col];
        // D = A(16x4 f32) * B(4x16 f32) + C  ->  v_wmma_f32_16x16x4_f32
        acc = __builtin_amdgcn_wmma_f32_16x16x4_f32(
            false, a, false, bf, (short)0, acc, false, false);
    }

    // C/D layout: VGPR r -> M = r + 8*half, N = lane&15.
    const float bias1 = b1[col];
    const float g2  = ln2_g[col];
    const float bt2 = ln2_b[col];
    const float w2  = W2[col];
    const float b2s = b2[0];

    #pragma unroll
    for (int r = 0; r < 8; ++r) {
        float h = fmaxf(acc[r] + bias1, 0.f);
        // LN2 over the 16-lane N dimension (xor masks 1..8 stay within a half)
        float s1 = h, s2 = h * h;
        #pragma unroll
        for (int m = 1; m < 16; m <<= 1) {
            s1 += __shfl_xor(s1, m, 32);
            s2 += __shfl_xor(s2, m, 32);
        }
        const float m2  = s1 * (1.0f / HID);
        const float v2  = s2 * (1.0f / HID) - m2 * m2;
        const float rs2 = rsqrtf(v2 + LN_EPS);
        const float h2  = (h - m2) * rs2 * g2 + bt2;
        float p = h2 * w2;
        #pragma unroll
        for (int m = 1; m < 16; m <<= 1) p += __shfl_xor(p, m, 32);
        if (col == 0) {
            scores[(size_t)batch * ROWS + rowbase + r + 8 * half] = p + b2s;
        }
    }
}

// ---------------------------------------------------------------------------
// Kernel 2: per batch -- min/max scale, in-LDS bitonic descending sort of
// theta with index payload (128 KB LDS), hard-rank bins, in-place PAVA
// isotonic regression, scatter of soft ranks.
// ---------------------------------------------------------------------------
__global__ __launch_bounds__(1024) void rank_kernel(
    float* __restrict__ scores_gs,   // [B,N] scores in; overwritten with sorted theta
    int*   __restrict__ perm,        // [B,N] descending-sort permutation
    float* __restrict__ out_soft, float* __restrict__ out_bins,
    const int* __restrict__ capacity_p)
{
    __shared__ float keys[N_];       // 64 KB
    __shared__ int   idx[N_];        // 64 KB
    __shared__ float redA[32];
    __shared__ float redB[32];
    __shared__ float s_mn, s_mx;

    const int tid = threadIdx.x;
    const int b   = blockIdx.x;
    const int cap = *capacity_p;
    float* row = scores_gs + (size_t)b * N_;

    // Load + block min/max (wave32 xor reductions, then cross-wave via LDS).
    float mn = INFINITY, mx = -INFINITY;
    for (int i = tid; i < N_; i += NT) {
        float v = row[i];
        keys[i] = v; idx[i] = i;
        mn = fminf(mn, v); mx = fmaxf(mx, v);
    }
    for (int m = 1; m < 32; m <<= 1) {
        mn = fminf(mn, __shfl_xor(mn, m, 32));
        mx = fmaxf(mx, __shfl_xor(mx, m, 32));
    }
    if ((tid & 31) == 0) { redA[tid >> 5] = mn; redB[tid >> 5] = mx; }
    __syncthreads();
    if (tid < 32) {
        mn = redA[tid]; mx = redB[tid];
        for (int m = 1; m < 32; m <<= 1) {
            mn = fminf(mn, __shfl_xor(mn, m, 32));
            mx = fmaxf(mx, __shfl_xor(mx, m, 32));
        }
        if (tid == 0) { s_mn = mn; s_mx = mx; }
    }
    __syncthreads();

    // theta = (s-mn)/(mx-mn)*100 / 0.01
    const float scale = 10000.0f / (s_mx - s_mn);
    for (int i = tid; i < N_; i += NT) keys[i] = (keys[i] - s_mn) * scale;
    __syncthreads();

    // Bitonic sort, descending, with index payload.
    for (int k = 2; k <= N_; k <<= 1) {
        for (int j = k >> 1; j > 0; j >>= 1) {
            for (int i = tid; i < N_; i += NT) {
                const int ixj = i ^ j;
                if (ixj > i) {
                    const float a = keys[i], c = keys[ixj];
                    const bool doSwap = ((i & k) == 0) ? (a < c) : (a > c);
                    if (doSwap) {
                        keys[i] = c; keys[ixj] = a;
                        const int q = idx[i]; idx[i] = idx[ixj]; idx[ixj] = q;
                    }
                }
            }
            __syncthreads();
        }
    }

    // Persist sorted theta + perm, emit bins (hard rank = N-1-i ascending),
    // then turn keys into y = theta_sorted - [N..1] for PAVA.
    for (int i = tid; i < N_; i += NT) {
        row[i] = keys[i];
        perm[(size_t)b * N_ + i] = idx[i];
        const int ra = N_ - 1 - i;
        out_bins[(size_t)b * N_ + idx[i]] = (float)(ra / cap + 1);
        keys[i] = keys[i] - (float)(N_ - i);
    }
    __syncthreads();

    // PAVA (non-increasing isotonic regression), stack reuses keys/idx in-place
    // (stack slot index <= current read index, so no clobbering).
    if (tid == 0) {
        int size = 0;
        for (int i = 0; i < N_; ++i) {
            const float y = keys[i];
            keys[size] = y; idx[size] = 1; ++size;
            while (size >= 2) {
                const float mp = keys[size - 2] / (float)idx[size - 2];
                const float mt = keys[size - 1] / (float)idx[size - 1];
                if (mp <= mt) {
                    keys[size - 2] += keys[size - 1];
                    idx[size - 2]  += idx[size - 1];
                    --size;
                } else break;
            }
        }
        // Expand block means; primal = theta_sorted - dual, scattered by perm.
        const int* gp = perm + (size_t)b * N_;
        int pos = 0;
        for (int j = 0; j < size; ++j) {
            const float meanv = keys[j] / (float)idx[j];
            const int cnt = idx[j];
            for (int tci = 0; tci < cnt; ++tci) {
                out_soft[(size_t)b * N_ + gp[pos]] = row[pos] - meanv;
                ++pos;
            }
        }
    }
}

// ---------------------------------------------------------------------------
extern "C" void kernel_launch(void* const* d_in, const int* in_sizes, int n_in,
                              void* d_out, int out_size, void* d_ws, size_t ws_size,
                              hipStream_t stream) {
    const float* table = (const float*)d_in[0];
    const float* ln1_g = (const float*)d_in[1];
    const float* ln1_b = (const float*)d_in[2];
    const float* W1    = (const float*)d_in[3];
    const float* b1    = (const float*)d_in[4];
    const float* ln2_g = (const float*)d_in[5];
    const float* ln2_b = (const float*)d_in[6];
    const float* W2    = (const float*)d_in[7];
    const float* b2    = (const float*)d_in[8];
    const int*   cap   = (const int*)d_in[9];

    // Workspace: scores[B*ROWS] f32 (reused as sorted theta), perm[B*ROWS] i32.
    float* scores = (float*)d_ws;
    int*   perm   = (int*)((char*)d_ws + (size_t)B_ * ROWS * sizeof(float));

    float* out_soft = (float*)d_out;                 // [B, ROWS, 1]
    float* out_bins = out_soft + (size_t)B_ * ROWS;  // [B, ROWS, 1] (as float)

    score_kernel<<<dim3(B_ * (ROWS / 64)), dim3(128), 0, stream>>>(
        table, ln1_g, ln1_b, W1, b1, ln2_g, ln2_b, W2, b2, scores);
    rank_kernel<<<dim3(B_), dim3(1024), 0, stream>>>(
        scores, perm, out_soft, out_bins, cap);
}